// TopKPooling_1_2731599200833
// MI455X (gfx1250) — compile-verified
//
#include <hip/hip_runtime.h>
#include <hip/hip_bf16.h>
#include <cstdint>

#define B_    16
#define S_    4096
#define D_    1024
#define K_    16
#define CHUNK 64   // d-columns per block (= blockDim.x); 2 waves/block
#define TS    64   // s-rows per LDS tile (tile = 64*64*4 = 16KB)

// ---------- CDNA5 (gfx1250) async global->LDS helpers ----------
// Async copy engine: per-lane 16B global->LDS, tracked by ASYNCcnt.
__device__ __forceinline__ void async_b128(unsigned lds_off, const void* gaddr) {
  asm volatile("global_load_async_to_lds_b128 %0, %1, off"
               :: "v"(lds_off), "v"(gaddr) : "memory");
}

template <int N>
__device__ __forceinline__ void wait_asynccnt() {
#if __has_builtin(__builtin_amdgcn_s_wait_asynccnt)
  __builtin_amdgcn_s_wait_asynccnt(N);
#else
  asm volatile("s_wait_asynccnt %0" :: "i"(N) : "memory");
#endif
}

// Copy a TS x CHUNK fp32 tile (row stride D_ in global) into LDS at lds_base.
// Per wave: 16 async B128 ops; lane covers (row = tid>>4, 16B col chunk).
__device__ __forceinline__ void prefetch_tile(const float* gbase, unsigned lds_base, int tid) {
  const int rowLane = tid >> 4;          // 0..3 within a 4-row group
  const int colB    = (tid & 15) << 4;   // byte col within 256B LDS row
  const char* g0 = (const char*)(gbase + (size_t)rowLane * D_) + colB;
  const unsigned l0 = lds_base + (unsigned)((rowLane << 8) + colB);
#pragma unroll
  for (int j = 0; j < 16; ++j) {         // 16 groups of 4 rows = 64 rows
    async_b128(l0 + (unsigned)(j << 10),                  // +1KB LDS per group
               g0 + (size_t)j * (4u * D_ * 4u));          // +4 rows in global
  }
}

__global__ __launch_bounds__(CHUNK) void topk_mean_kernel(
    const float* __restrict__ x, const int* __restrict__ mask, float* __restrict__ out)
{
  __shared__ float tile[2][TS * CHUNK];  // 32KB double buffer
  __shared__ int s_len;

  const int tid = threadIdx.x;
  const int b  = blockIdx.x >> 4;              // D_/CHUNK = 16 chunks per batch
  const int d0 = (blockIdx.x & 15) * CHUNK;

  // Mask is a prefix of ones: binary-search its length once per block.
  if (tid == 0) {
    const int* m = mask + (size_t)b * S_;
    int lo = 0, hi = S_;
    while (lo < hi) { int mid = (lo + hi) >> 1; if (m[mid] != 0) lo = mid + 1; else hi = mid; }
    s_len = lo;                                // guaranteed >= K_
  }
  __syncthreads();
  const int len    = s_len;
  const int ntiles = (len + TS - 1) / TS;      // tail rows stay inside x (len <= S_)

  const float* gb = x + (size_t)b * S_ * D_ + d0;
  const unsigned lds0 = (unsigned)(uintptr_t)(void*)&tile[0][0];
  const unsigned lds1 = (unsigned)(uintptr_t)(void*)&tile[1][0];

  // Per-lane top-16 set (unsorted) + conservative lower bound on its min.
  float tk[K_];
#pragma unroll
  for (int i = 0; i < K_; ++i) tk[i] = -3.0e38f;
  float cmin = -3.0e38f;

  prefetch_tile(gb, lds0, tid);                // tile 0 in flight

  for (int t = 0; t < ntiles; ++t) {
    const int s0 = t * TS;
    const bool more = (t + 1) < ntiles;
    if (more) prefetch_tile(gb + (size_t)(s0 + TS) * D_, (t & 1) ? lds0 : lds1, tid);
    if (more) wait_asynccnt<16>();             // in-order: current tile's 16 ops done
    else      wait_asynccnt<0>();
    __syncthreads();                           // all waves' copies visible

    const float* lbuf = (t & 1) ? &tile[1][0] : &tile[0][0];
    const int rows = min(TS, len - s0);
#pragma unroll 4
    for (int r = 0; r < rows; ++r) {
      const float v = lbuf[r * CHUNK + tid];
      if (v > cmin) {
        // Bubble v through the 16-entry set: min of the 17 falls out as w.
        float w = v;
#pragma unroll
        for (int i = 0; i < K_; ++i) {
          const float mn = fminf(tk[i], w);
          const float mx = fmaxf(tk[i], w);
          tk[i] = mx;
          w = mn;
        }
        cmin = w;                              // <= exact min of kept set (safe)
      }
    }
    __syncthreads();                           // done reading before overwrite
  }

  float s = 0.f;
#pragma unroll
  for (int i = 0; i < K_; ++i) s += tk[i];
  out[(size_t)b * D_ + d0 + tid] = s * (1.0f / (float)K_);
}

extern "C" void kernel_launch(void* const* d_in, const int* in_sizes, int n_in,
                              void* d_out, int out_size, void* d_ws, size_t ws_size,
                              hipStream_t stream) {
  (void)in_sizes; (void)n_in; (void)d_ws; (void)ws_size; (void)out_size;
  const float* x    = (const float*)d_in[0];
  const int*   mask = (const int*)d_in[1];
  float*       out  = (float*)d_out;
  dim3 grid(B_ * (D_ / CHUNK));   // 256 blocks
  dim3 block(CHUNK);              // 64 threads = 2 wave32
  hipLaunchKernelGGL(topk_mean_kernel, grid, block, 0, stream, x, mask, out);
}